// NonSymSimplePGC_83519934038261
// MI455X (gfx1250) — compile-verified
//
#include <hip/hip_runtime.h>
#include <hip/hip_bf16.h>
#include <math.h>

// ---------------------------------------------------------------------------
// Problem constants (from the reference)
// ---------------------------------------------------------------------------
#define NPARTS 16
#define DDIM   16
#define EPSV   1e-8f

typedef __attribute__((ext_vector_type(2))) float v2f;
typedef __attribute__((ext_vector_type(8))) float v8f;

union V8U { v8f v; float f[8]; };

// ---------------------------------------------------------------------------
// Serial 16x16 slogdet via LU with partial pivoting.
// Returns logabs(det) + log(sign)  (log(-1)=NaN, matching reference semantics).
// ---------------------------------------------------------------------------
__device__ __forceinline__ float slogdet16(float m[DDIM][DDIM]) {
    float sign = 1.0f, logabs = 0.0f;
    for (int k = 0; k < DDIM; ++k) {
        int piv = k;
        float best = fabsf(m[k][k]);
        for (int r = k + 1; r < DDIM; ++r) {
            float v = fabsf(m[r][k]);
            if (v > best) { best = v; piv = r; }
        }
        if (piv != k) {
            for (int j = 0; j < DDIM; ++j) {
                float t = m[k][j]; m[k][j] = m[piv][j]; m[piv][j] = t;
            }
            sign = -sign;
        }
        float pk = m[k][k];
        if (pk < 0.0f) sign = -sign;
        logabs += logf(fabsf(pk));
        float inv = 1.0f / pk;
        for (int r = k + 1; r < DDIM; ++r) {
            float f = m[r][k] * inv;
            for (int j = k + 1; j < DDIM; ++j) m[r][j] -= f * m[k][j];
        }
    }
    return logabs + logf(sign);
}

// ---------------------------------------------------------------------------
// Kernel 1: one wave32. Builds L0 = A^T A + B C^T - C B^T + eps*I with
// V_WMMA_F32_16X16X4_F32 (f32 matrix pipe), and the 16x16 log-softmax table.
//
// WMMA f32 16x16x4 operand layout (ISA 7.12.2):
//   A-operand (16x4, MxK):  element(m,kk) -> lane = m (0..15) in each half;
//                           kk = 2*(lane>>4) + vgpr_r
//   B-operand (4x16, KxN):  element(kk,n) -> lane = n (halves split kk);
//                           kk = 2*(lane>>4) + vgpr_r
//   C/D (16x16): element(M,N) -> vgpr d, lane: M = d + 8*(lane>>4), N = lane&15
// ---------------------------------------------------------------------------
__global__ __launch_bounds__(32)
void pgc_prep_kernel(const float* __restrict__ A, const float* __restrict__ B,
                     const float* __restrict__ C, const float* __restrict__ W,
                     float* __restrict__ L0out, float* __restrict__ logw)
{
    const int lane = threadIdx.x;      // 0..31, EXEC all ones for WMMA
    const int col  = lane & 15;
    const int half = lane >> 4;        // 0 or 1

    V8U acc; acc.v = (v8f){};

    // A^T A :  Amat[m,kk] = A[k0+kk, m],  Bmat[kk,n] = A[k0+kk, n]
    #pragma unroll
    for (int k0 = 0; k0 < DDIM; k0 += 4) {
        v2f a, b;
        a.x = A[(k0 + 2 * half + 0) * DDIM + col];
        a.y = A[(k0 + 2 * half + 1) * DDIM + col];
        b.x = a.x; b.y = a.y;
        acc.v = __builtin_amdgcn_wmma_f32_16x16x4_f32(
            false, a, false, b, (short)0, acc.v, false, false);
    }
    // + B C^T :  Amat[m,kk] = B[m, k0+kk],  Bmat[kk,n] = C[n, k0+kk]
    #pragma unroll
    for (int k0 = 0; k0 < DDIM; k0 += 4) {
        v2f a, b;
        a.x = B[col * DDIM + k0 + 2 * half + 0];
        a.y = B[col * DDIM + k0 + 2 * half + 1];
        b.x = C[col * DDIM + k0 + 2 * half + 0];
        b.y = C[col * DDIM + k0 + 2 * half + 1];
        acc.v = __builtin_amdgcn_wmma_f32_16x16x4_f32(
            false, a, false, b, (short)0, acc.v, false, false);
    }
    // - C B^T :  Amat[m,kk] = -C[m, k0+kk],  Bmat[kk,n] = B[n, k0+kk]
    // (f32 WMMA has no A/B NEG modifier -> negate operand manually)
    #pragma unroll
    for (int k0 = 0; k0 < DDIM; k0 += 4) {
        v2f a, b;
        a.x = -C[col * DDIM + k0 + 2 * half + 0];
        a.y = -C[col * DDIM + k0 + 2 * half + 1];
        b.x = B[col * DDIM + k0 + 2 * half + 0];
        b.y = B[col * DDIM + k0 + 2 * half + 1];
        acc.v = __builtin_amdgcn_wmma_f32_16x16x4_f32(
            false, a, false, b, (short)0, acc.v, false, false);
    }

    // + eps*I and write out L0 (row = d + 8*half, col = lane&15)
    #pragma unroll
    for (int d = 0; d < 8; ++d) {
        int row = d + 8 * half;
        float v = acc.f[d] + ((row == col) ? EPSV : 0.0f);
        L0out[row * DDIM + col] = v;
    }

    // log-softmax table: logw[p][0] = 0 (gated slot), logw[p][1+k] = logsoftmax(W[p])[k]
    if (lane < NPARTS) {
        float vals[15];
        float mx = -INFINITY;
        #pragma unroll
        for (int k = 0; k < 15; ++k) {
            vals[k] = W[lane * 15 + k];
            mx = fmaxf(mx, vals[k]);
        }
        float s = 0.0f;
        #pragma unroll
        for (int k = 0; k < 15; ++k) s += expf(vals[k] - mx);
        float lse = mx + logf(s);
        logw[lane * 16 + 0] = 0.0f;
        #pragma unroll
        for (int k = 0; k < 15; ++k) logw[lane * 16 + 1 + k] = vals[k] - lse;
    }
}

// ---------------------------------------------------------------------------
// Kernel 2a: alpha = slogdet(L0 + I)  (single scalar)
// ---------------------------------------------------------------------------
__global__ void pgc_alpha_kernel(const float* __restrict__ L0, float* __restrict__ alpha)
{
    if (threadIdx.x == 0 && blockIdx.x == 0) {
        float m[DDIM][DDIM];
        for (int i = 0; i < DDIM; ++i)
            for (int j = 0; j < DDIM; ++j)
                m[i][j] = L0[i * DDIM + j] + ((i == j) ? 1.0f : 0.0f);
        alpha[0] = slogdet16(m);
    }
}

// ---------------------------------------------------------------------------
// Kernel 2b: sub-determinant table. For every 16-bit mask S:
//   table[S] = slogdet of L0 with columns j not in S replaced by e_j
//            = log|det(L0[S,S])| + log(sign)
// 65536 independent threads, tiny total FLOPs.
// ---------------------------------------------------------------------------
__global__ __launch_bounds__(256)
void pgc_table_kernel(const float* __restrict__ L0, float* __restrict__ table)
{
    unsigned mask = blockIdx.x * blockDim.x + threadIdx.x;   // 0..65535
    float m[DDIM][DDIM];
    for (int i = 0; i < DDIM; ++i) {
        for (int j = 0; j < DDIM; ++j) {
            bool inS = (mask >> j) & 1u;
            m[i][j] = inS ? L0[i * DDIM + j] : ((i == j) ? 1.0f : 0.0f);
        }
    }
    table[mask] = slogdet16(m);
}

// ---------------------------------------------------------------------------
// Kernel 3: bandwidth-bound batch pass. 16 lanes per row; each lane loads one
// int4 (4 bit-flags -> idx), gathers log-weight from LDS, builds the 16-bit
// subset mask, shuffle-reduces within the 16-lane group, then one table gather.
// Coalesced: thread t of a block reads int4 #(blockBase*16 + t).
// ---------------------------------------------------------------------------
__global__ __launch_bounds__(256)
void pgc_batch_kernel(const int* __restrict__ x, const float* __restrict__ logw,
                      const float* __restrict__ table, const float* __restrict__ alpha,
                      float* __restrict__ out, int batch)
{
    __shared__ float lw[256];
    int t = threadIdx.x;
    lw[t] = logw[t];
    __syncthreads();

    int gtid   = blockIdx.x * blockDim.x + t;
    int row    = gtid >> 4;
    int part   = t & 15;
    if (row >= batch) return;

    const int4* xp = (const int4*)x + (size_t)row * 16 + part;
    // Prefetch a few KB ahead of the streaming read (global_prefetch_b8)
    __builtin_prefetch((const void*)(xp + 1024), 0, 0);
    int4 b = *xp;

    int idx = b.x | (b.y << 1) | (b.z << 2) | (b.w << 3);
    float contrib = (idx == 0) ? 0.0f : lw[part * 16 + idx];
    unsigned mbit = (idx != 0) ? (1u << part) : 0u;

    // reduce over the 16-lane group
    #pragma unroll
    for (int off = 8; off > 0; off >>= 1) {
        contrib += __shfl_xor(contrib, off, 16);
        mbit    |= __shfl_xor(mbit, off, 16);
    }

    if (part == 0) {
        out[row] = table[mbit] + contrib - alpha[0];
    }
}

// ---------------------------------------------------------------------------
// Launch
// ---------------------------------------------------------------------------
extern "C" void kernel_launch(void* const* d_in, const int* in_sizes, int n_in,
                              void* d_out, int out_size, void* d_ws, size_t ws_size,
                              hipStream_t stream)
{
    const int*   x = (const int*)d_in[0];     // (BATCH, 64) int32
    const float* W = (const float*)d_in[1];   // (16, 15)
    const float* A = (const float*)d_in[2];   // (16, 16)
    const float* B = (const float*)d_in[3];   // (16, 16)
    const float* C = (const float*)d_in[4];   // (16, 16)

    float* ws    = (float*)d_ws;
    float* L0    = ws;            // 256 floats
    float* logw  = ws + 256;      // 256 floats
    float* alpha = ws + 512;      // 1 float (pad to 1024)
    float* table = ws + 1024;     // 65536 floats (256 KB)

    int batch = in_sizes[0] / 64;

    pgc_prep_kernel<<<1, 32, 0, stream>>>(A, B, C, W, L0, logw);
    pgc_alpha_kernel<<<1, 32, 0, stream>>>(L0, alpha);
    pgc_table_kernel<<<65536 / 256, 256, 0, stream>>>(L0, table);

    int threads = batch * 16;
    pgc_batch_kernel<<<(threads + 255) / 256, 256, 0, stream>>>(
        x, logw, table, alpha, (float*)d_out, batch);
}